// Relative_AttentionHead_7799660610311
// MI455X (gfx1250) — compile-verified
//
#include <hip/hip_runtime.h>
#include <math.h>
#include <stdint.h>

// B=4, L=1024, D=512, KD=64, H=32
#define NB 4
#define LL 1024
#define DD 512
#define KDIM 64
#define HH 32
#define NCHUNK 4           // k-split for flash-decoding style parallelism
#define CHUNK (LL / NCHUNK)
#define NTILES (CHUNK / 16)

typedef float v2f __attribute__((ext_vector_type(2)));
typedef float v8f __attribute__((ext_vector_type(8)));
typedef int   v4i __attribute__((vector_size(16)));   // matches builtin param type

#define NEG_INF (-__builtin_inff())

// ---- CDNA5 async global->LDS path (guarded; sync fallback otherwise) ------
#if __has_builtin(__builtin_amdgcn_global_load_async_to_lds_b128) && \
    __has_builtin(__builtin_amdgcn_s_wait_asynccnt)
#define USE_ASYNC_LDS 1
#else
#define USE_ASYNC_LDS 0
#endif

__device__ __forceinline__ void async_copy16(const void* g, void* l) {
#if USE_ASYNC_LDS
    __builtin_amdgcn_global_load_async_to_lds_b128(
        (__attribute__((address_space(1))) v4i*)(uintptr_t)g,
        (__attribute__((address_space(3))) v4i*)(uint32_t)(uintptr_t)l,
        0, 0);
#else
    *(float4*)l = *(const float4*)g;
#endif
}

#if USE_ASYNC_LDS
#define WAIT_ASYNC(n) __builtin_amdgcn_s_wait_asynccnt(n)
#else
#define WAIT_ASYNC(n)
#endif

__device__ __forceinline__ float gelu_f(float x) {
    // exact (erf-based) GELU, matching jax.nn.gelu(approximate=False)
    return 0.5f * x * (1.0f + erff(x * 0.70710678118654752f));
}

__device__ __forceinline__ float bfly_max16(float x) {
#pragma unroll
    for (int m = 1; m < 16; m <<= 1) x = fmaxf(x, __shfl_xor(x, m, 32));
    return x;
}
__device__ __forceinline__ float bfly_sum16(float x) {
#pragma unroll
    for (int m = 1; m < 16; m <<= 1) x += __shfl_xor(x, m, 32);
    return x;
}

// ---------------------------------------------------------------------------
// Kernel A: Q/K/V projection.  One wave per 16-row M-tile of src (256 tiles).
// GEMM: [16 x 512] x [512 x 64] x 3 matrices, f32 WMMA 16x16x4, K-loop 128.
// A-frag layout (16x4 f32): lanes 0-15 -> K=0,1 ; lanes 16-31 -> K=2,3.
// ---------------------------------------------------------------------------
__global__ __launch_bounds__(32) void qkv_kernel(
    const float* __restrict__ src,
    const float* __restrict__ Wq, const float* __restrict__ Wk,
    const float* __restrict__ Wv,
    float* __restrict__ Qo, float* __restrict__ Ko, float* __restrict__ Vo)
{
    const int mt    = blockIdx.x;        // 0..255 (16 rows of B*L each)
    const int lane  = threadIdx.x;
    const int rlo   = lane & 15;
    const int khalf = lane >> 4;

    const float* W[3]   = {Wq, Wk, Wv};
    float*       Out[3] = {Qo, Ko, Vo};

    v8f acc[3][4];
#pragma unroll
    for (int m = 0; m < 3; m++)
#pragma unroll
        for (int nt = 0; nt < 4; nt++) acc[m][nt] = (v8f)0.0f;

    const float* srcRow = src + (size_t)(mt * 16 + rlo) * DD;
    for (int s = 0; s < DD / 4; s++) {
        v2f a = *(const v2f*)(srcRow + s * 4 + 2 * khalf);
#pragma unroll
        for (int m = 0; m < 3; m++) {
#pragma unroll
            for (int nt = 0; nt < 4; nt++) {
                v2f bf = *(const v2f*)(W[m] + (size_t)(nt * 16 + rlo) * DD + s * 4 + 2 * khalf);
                acc[m][nt] = __builtin_amdgcn_wmma_f32_16x16x4_f32(
                    false, a, false, bf, (short)0, acc[m][nt], false, false);
            }
        }
    }
#pragma unroll
    for (int m = 0; m < 3; m++)
#pragma unroll
        for (int nt = 0; nt < 4; nt++)
#pragma unroll
            for (int v = 0; v < 8; v++) {
                int row = khalf ? (v + 8) : v;
                Out[m][(size_t)(mt * 16 + row) * KDIM + nt * 16 + rlo] = acc[m][nt][v];
            }
}

// ---------------------------------------------------------------------------
// Kernel B: fused bias-MLP + flash attention over one k-chunk.
// Block = 4 waves; wave w owns 16 q rows.  grid = B * (L/64) * NCHUNK = 256.
// K/V tiles: double-buffered async global->LDS copies (ASYNCcnt pipeline).
// Per 16-k tile: MLP bias tile (layer1 VALU in A-frag layout, layer2 WMMA,
// layer3 butterfly), QK^T WMMA + bias + scale + mask, online softmax, P@V WMMA.
// ---------------------------------------------------------------------------
__global__ __launch_bounds__(128) void attn_kernel(
    const float* __restrict__ bias_tensor,      // [B,L,L,2]
    const unsigned char* __restrict__ mask,     // [B,L,L] bool
    const float* __restrict__ mult,             // [B]
    const float* __restrict__ sp,               // [1]
    const float* __restrict__ w1,               // [32,2]
    const float* __restrict__ b1,               // [32]
    const float* __restrict__ w2,               // [32,32]
    const float* __restrict__ b2,               // [32]
    const float* __restrict__ w3,               // [1,32]
    const float* __restrict__ b3,               // [1]
    const float* __restrict__ Qw,               // [B*L,64]
    const float* __restrict__ Kw,
    const float* __restrict__ Vw,
    float* __restrict__ Opart,                  // [NCHUNK, B*L, 64]
    float* __restrict__ Mpart,                  // [NCHUNK, B*L]
    float* __restrict__ Lpart)                  // [NCHUNK, B*L]
{
    __shared__ float Kt[2][16 * KDIM];
    __shared__ float Vt[2][16 * KDIM];
    __shared__ float BiasT[4][256];
    __shared__ float Pt[4][256];

    const int bx = blockIdx.x;
    const int ch = bx & 3;
    const int qt = (bx >> 2) & 15;
    const int b  = bx >> 6;

    const int w     = threadIdx.x >> 5;
    const int lane  = threadIdx.x & 31;
    const int rlo   = lane & 15;
    const int khalf = lane >> 4;
    const int q0w   = qt * 64 + w * 16;
    const int k0base = ch * CHUNK;

    // issue one K/V tile copy into LDS buffer `buf` (4 async ops per wave)
    auto issue_tile = [&](int buf, int k0) {
        const float4* Ks = (const float4*)(Kw + ((size_t)b * LL + k0) * KDIM);
        const float4* Vs = (const float4*)(Vw + ((size_t)b * LL + k0) * KDIM);
        for (int i = threadIdx.x; i < 16 * KDIM / 4; i += 128) {
            async_copy16(Ks + i, &Kt[buf][4 * i]);
            async_copy16(Vs + i, &Vt[buf][4 * i]);
        }
    };

    // ---- per-wave preloads (wave-invariant weights into registers) ----
    issue_tile(0, k0base);              // prologue copy for tile 0

    v2f qa[16];
    const float* Qrow = Qw + ((size_t)b * LL + q0w + rlo) * KDIM;
#pragma unroll
    for (int s = 0; s < 16; s++) qa[s] = *(const v2f*)(Qrow + s * 4 + 2 * khalf);

    v2f w2f0[8], w2f1[8];
#pragma unroll
    for (int s = 0; s < 8; s++) {
        w2f0[s] = *(const v2f*)(w2 + (size_t)rlo * HH + s * 4 + 2 * khalf);
        w2f1[s] = *(const v2f*)(w2 + (size_t)(16 + rlo) * HH + s * 4 + 2 * khalf);
    }
    v2f w1lo[8], w1hi[8];
    float b1lo[8], b1hi[8];
#pragma unroll
    for (int s = 0; s < 8; s++) {
        int h0 = s * 4 + 2 * khalf;
        w1lo[s] = *(const v2f*)(w1 + h0 * 2);
        w1hi[s] = *(const v2f*)(w1 + (h0 + 1) * 2);
        b1lo[s] = b1[h0];
        b1hi[s] = b1[h0 + 1];
    }
    const float b2a = b2[rlo],      b2b = b2[16 + rlo];
    const float w3a = w3[rlo],      w3b = w3[16 + rlo];
    const float b3s = b3[0];
    const float scale = sp[0] * mult[b] * 0.125f;   // /sqrt(64) * s * multiplier

    v8f Oacc[4];
#pragma unroll
    for (int nt = 0; nt < 4; nt++) Oacc[nt] = (v8f)0.0f;
    float rowm[8], rowl[8];
#pragma unroll
    for (int v = 0; v < 8; v++) { rowm[v] = NEG_INF; rowl[v] = 0.0f; }

    for (int kt = 0; kt < NTILES; kt++) {
        const int k0  = k0base + kt * 16;
        const int buf = kt & 1;

        // ---- start next tile's async copy, then wait for current tile ----
        if (kt + 1 < NTILES) {
            issue_tile(buf ^ 1, k0 + 16);
            // prefetch next bias_tensor tile rows (streamed exactly once)
            __builtin_prefetch(
                bias_tensor + (((size_t)b * LL + q0w + rlo) * LL + (k0 + 16)) * 2, 0, 0);
            WAIT_ASYNC(4);   // 4 newest ops belong to the next buffer
        } else {
            WAIT_ASYNC(0);
        }
        __syncthreads();

        // ---- bias MLP: 16 q rows x 16 k cols (one 16x16 tile per mt) ----
#pragma unroll 2
        for (int mt = 0; mt < 16; mt++) {
            const int q = q0w + mt;
            v2f x = *(const v2f*)(bias_tensor +
                                  (((size_t)b * LL + q) * LL + (k0 + rlo)) * 2);
            v8f c0 = (v8f)0.0f, c1 = (v8f)0.0f;
#pragma unroll
            for (int s = 0; s < 8; s++) {
                // layer 1 + GELU, computed directly in A-fragment layout
                float a0 = gelu_f(fmaf(x[0], w1lo[s][0], fmaf(x[1], w1lo[s][1], b1lo[s])));
                float a1 = gelu_f(fmaf(x[0], w1hi[s][0], fmaf(x[1], w1hi[s][1], b1hi[s])));
                v2f af; af[0] = a0; af[1] = a1;
                c0 = __builtin_amdgcn_wmma_f32_16x16x4_f32(false, af, false, w2f0[s], (short)0, c0, false, false);
                c1 = __builtin_amdgcn_wmma_f32_16x16x4_f32(false, af, false, w2f1[s], (short)0, c1, false, false);
            }
            float tv[8];
#pragma unroll
            for (int v = 0; v < 8; v++) {
                float h2a = gelu_f(c0[v] + b2a);
                float h2b = gelu_f(c1[v] + b2b);
                tv[v] = fmaf(h2a, w3a, h2b * w3b);
            }
#pragma unroll
            for (int v = 0; v < 8; v++) tv[v] = bfly_sum16(tv[v]);
            if (rlo == 0) {
#pragma unroll
                for (int v = 0; v < 8; v++)
                    BiasT[w][mt * 16 + (khalf ? v + 8 : v)] = tv[v] + b3s;
            }
        }

        // ---- S = Q K^T (16x16, K=64) ----
        v8f sacc = (v8f)0.0f;
#pragma unroll
        for (int s = 0; s < 16; s++) {
            v2f kf = *(const v2f*)(&Kt[buf][rlo * KDIM + s * 4 + 2 * khalf]);
            sacc = __builtin_amdgcn_wmma_f32_16x16x4_f32(false, qa[s], false, kf, (short)0, sacc, false, false);
        }
#pragma unroll
        for (int v = 0; v < 8; v++) {
            const int r = khalf ? v + 8 : v;
            float sv = (sacc[v] + BiasT[w][r * 16 + rlo]) * scale;
            if (mask[((size_t)b * LL + (q0w + r)) * LL + (k0 + rlo)]) sv = NEG_INF;
            sacc[v] = sv;
        }

        // ---- online softmax update ----
#pragma unroll
        for (int v = 0; v < 8; v++) {
            float mcur  = bfly_max16(sacc[v]);
            float mnew  = fmaxf(rowm[v], mcur);
            float alpha = __expf(rowm[v] - mnew);
            float p     = __expf(sacc[v] - mnew);
            float rs    = bfly_sum16(p);
            rowl[v] = fmaf(rowl[v], alpha, rs);
            rowm[v] = mnew;
            sacc[v] = p;
#pragma unroll
            for (int nt = 0; nt < 4; nt++) Oacc[nt][v] *= alpha;
        }

        // ---- P through LDS (C-layout -> A-layout), then O += P @ V ----
#pragma unroll
        for (int v = 0; v < 8; v++)
            Pt[w][(khalf ? v + 8 : v) * 16 + rlo] = sacc[v];
#pragma unroll
        for (int t = 0; t < 4; t++) {
            v2f pf = *(const v2f*)(Pt[w] + rlo * 16 + t * 4 + 2 * khalf);
#pragma unroll
            for (int nt = 0; nt < 4; nt++) {
                v2f vf;
                vf[0] = Vt[buf][(t * 4 + 2 * khalf + 0) * KDIM + nt * 16 + rlo];
                vf[1] = Vt[buf][(t * 4 + 2 * khalf + 1) * KDIM + nt * 16 + rlo];
                Oacc[nt] = __builtin_amdgcn_wmma_f32_16x16x4_f32(false, pf, false, vf, (short)0, Oacc[nt], false, false);
            }
        }
        __syncthreads();   // all waves done with buffer before it is re-filled
    }

    // ---- write chunk partials ----
    float* Orow = Opart + ((size_t)ch * (NB * LL) + (size_t)b * LL + q0w) * KDIM;
#pragma unroll
    for (int nt = 0; nt < 4; nt++)
#pragma unroll
        for (int v = 0; v < 8; v++) {
            int r = khalf ? v + 8 : v;
            Orow[(size_t)r * KDIM + nt * 16 + rlo] = Oacc[nt][v];
        }
    if (rlo == 0) {
#pragma unroll
        for (int v = 0; v < 8; v++) {
            int r = khalf ? v + 8 : v;
            Mpart[(size_t)ch * (NB * LL) + b * LL + q0w + r] = rowm[v];
            Lpart[(size_t)ch * (NB * LL) + b * LL + q0w + r] = rowl[v];
        }
    }
}

// ---------------------------------------------------------------------------
// Kernel C: combine the NCHUNK flash partials per row.
// ---------------------------------------------------------------------------
__global__ __launch_bounds__(256) void combine_kernel(
    const float* __restrict__ Opart, const float* __restrict__ Mpart,
    const float* __restrict__ Lpart, float* __restrict__ out)
{
    const int idx = blockIdx.x * 256 + threadIdx.x;   // B*L*64 threads
    const int row = idx >> 6;
    const int d   = idx & 63;

    float mmax = NEG_INF;
#pragma unroll
    for (int ch = 0; ch < NCHUNK; ch++)
        mmax = fmaxf(mmax, Mpart[ch * (NB * LL) + row]);

    float num = 0.0f, den = 0.0f;
#pragma unroll
    for (int ch = 0; ch < NCHUNK; ch++) {
        float e = __expf(Mpart[ch * (NB * LL) + row] - mmax);
        num = fmaf(Opart[((size_t)ch * (NB * LL) + row) * KDIM + d], e, num);
        den = fmaf(Lpart[ch * (NB * LL) + row], e, den);
    }
    out[(size_t)row * KDIM + d] = num / den;
}

// ---------------------------------------------------------------------------
extern "C" void kernel_launch(void* const* d_in, const int* in_sizes, int n_in,
                              void* d_out, int out_size, void* d_ws, size_t ws_size,
                              hipStream_t stream)
{
    (void)in_sizes; (void)n_in; (void)out_size; (void)ws_size;

    const float*         src         = (const float*)d_in[0];
    const float*         bias_tensor = (const float*)d_in[1];
    const unsigned char* mask        = (const unsigned char*)d_in[2];
    const float*         mult        = (const float*)d_in[3];
    const float*         Wq          = (const float*)d_in[4];
    const float*         Wk          = (const float*)d_in[5];
    const float*         Wv          = (const float*)d_in[6];
    const float*         sp          = (const float*)d_in[7];
    const float*         w1          = (const float*)d_in[8];
    const float*         b1          = (const float*)d_in[9];
    const float*         w2          = (const float*)d_in[10];
    const float*         b2          = (const float*)d_in[11];
    const float*         w3          = (const float*)d_in[12];
    const float*         b3          = (const float*)d_in[13];
    float*               out         = (float*)d_out;

    // workspace layout (floats): Q | K | V | Opart | Mpart | Lpart  (~7.3 MB)
    float* Qw = (float*)d_ws;
    float* Kw = Qw + (size_t)NB * LL * KDIM;
    float* Vw = Kw + (size_t)NB * LL * KDIM;
    float* Op = Vw + (size_t)NB * LL * KDIM;
    float* Mp = Op + (size_t)NCHUNK * NB * LL * KDIM;
    float* Lp = Mp + (size_t)NCHUNK * NB * LL;

    hipLaunchKernelGGL(qkv_kernel, dim3((NB * LL) / 16), dim3(32), 0, stream,
                       src, Wq, Wk, Wv, Qw, Kw, Vw);

    hipLaunchKernelGGL(attn_kernel, dim3(NB * (LL / 64) * NCHUNK), dim3(128), 0, stream,
                       bias_tensor, mask, mult, sp, w1, b1, w2, b2, w3, b3,
                       Qw, Kw, Vw, Op, Mp, Lp);

    hipLaunchKernelGGL(combine_kernel, dim3((NB * LL * KDIM) / 256), dim3(256), 0, stream,
                       Op, Mp, Lp, out);
}